// GraphConv_23003844838036
// MI455X (gfx1250) — compile-verified
//
#include <hip/hip_runtime.h>

#define N_NODES 100000
#define N_EDGES 1600000
#define D 128
#define LDS_STRIDE 132   // 128 + 4 pad: keeps float2 fragment loads bank-spread

typedef __attribute__((ext_vector_type(2))) float v2f;
typedef __attribute__((ext_vector_type(8))) float v8f;

// Non-returning device-scope f32 atomic add (GLOBAL_ATOMIC_ADD_F32, STOREcnt).
__device__ __forceinline__ void atomic_add_f32_dev(float* p, float v) {
    asm volatile("global_atomic_add_f32 %0, %1, off scope:SCOPE_DEV"
                 :: "v"(p), "v"(v) : "memory");
}

// ---------------------------------------------------------------------------
// Kernel 1: zero the h_neigh accumulator in workspace (grid-stride float4).
// ---------------------------------------------------------------------------
__global__ void gc_zero_kernel(float4* __restrict__ ws, int n4) {
    int i = blockIdx.x * blockDim.x + threadIdx.x;
    int stride = gridDim.x * blockDim.x;
    float4 z = {0.f, 0.f, 0.f, 0.f};
    for (; i < n4; i += stride) ws[i] = z;
}

// ---------------------------------------------------------------------------
// Kernel 2: edge scatter. One wave (32 lanes) per edge; each lane moves a
// float4 (coalesced 512B row) and issues 4 L2-resident f32 atomics.
// ---------------------------------------------------------------------------
__global__ void gc_scatter_kernel(const float* __restrict__ feat,
                                  const int*   __restrict__ src,
                                  const int*   __restrict__ dst,
                                  float*       __restrict__ h) {
    long gid  = (long)blockIdx.x * blockDim.x + threadIdx.x;
    long e    = gid >> 5;
    int  lane = (int)(gid & 31);
    if (e >= N_EDGES) return;

    int s = src[e];
    int d = dst[e];

    const float4 f = *(const float4*)(feat + (long)s * D + lane * 4);
    float* hp = h + (long)d * D + lane * 4;
    atomic_add_f32_dev(hp + 0, f.x);
    atomic_add_f32_dev(hp + 1, f.y);
    atomic_add_f32_dev(hp + 2, f.z);
    atomic_add_f32_dev(hp + 3, f.w);
}

// ---------------------------------------------------------------------------
// Kernel 3: out = h @ W^T + b via V_WMMA_F32_16X16X4_F32.
// Block = 256 threads (8 waves). Block computes 64 rows x 128 cols.
// Wave w owns N-tile [16w, 16w+16); iterates 4 M-tiles x 32 K-steps.
// W (128x128) and a 64x128 h tile staged in LDS (padded stride 132).
// ---------------------------------------------------------------------------
__launch_bounds__(256)
__global__ void gc_gemm_kernel(const float* __restrict__ h,
                               const float* __restrict__ W,
                               const float* __restrict__ bias,
                               float*       __restrict__ out) {
    __shared__ float sW[D * LDS_STRIDE];   // sW[o*132 + k] = W[o][k]
    __shared__ float sA[64 * LDS_STRIDE];  // sA[m*132 + k] = h[row0+m][k]

    const int tid  = threadIdx.x;
    const int row0 = blockIdx.x * 64;

    // Stage W: 128x128 = 4096 float4 units, 16 per thread, coalesced.
    for (int i = tid; i < (D * D) / 4; i += 256) {
        int r  = i >> 5;     // row (32 float4 per row)
        int c4 = i & 31;
        float4 w = *(const float4*)(W + r * D + c4 * 4);
        *(float4*)(&sW[r * LDS_STRIDE + c4 * 4]) = w;
    }
    // Stage A tile: 64x128 with row bounds check (zero-fill past N_NODES).
    for (int i = tid; i < (64 * D) / 4; i += 256) {
        int r  = i >> 5;
        int c4 = i & 31;
        float4 a = {0.f, 0.f, 0.f, 0.f};
        if (row0 + r < N_NODES)
            a = *(const float4*)(h + (long)(row0 + r) * D + c4 * 4);
        *(float4*)(&sA[r * LDS_STRIDE + c4 * 4]) = a;
    }
    __syncthreads();

    const int wave  = tid >> 5;          // 0..7 -> N tile
    const int lane  = tid & 31;
    const int n     = lane & 15;         // column within tile (and A row idx)
    const int khalf = (lane < 16) ? 0 : 2;  // ISA 16x4 f32 layout: K pair select
    const int nbase = wave * 16;
    const float bval = bias[nbase + n];  // each lane's 8 acc elems share N

    for (int mt = 0; mt < 4; ++mt) {
        const int mbase = mt * 16;
        v8f acc = {};
        #pragma unroll
        for (int k = 0; k < D; k += 4) {
            // A fragment: lane holds h[row0+mbase+n][k+khalf .. +1]
            v2f a = *(const v2f*)(&sA[(mbase + n) * LDS_STRIDE + k + khalf]);
            // B fragment: B[k'][n] = W[nbase+n][k+k'] -> contiguous float2
            v2f b = *(const v2f*)(&sW[(nbase + n) * LDS_STRIDE + k + khalf]);
            acc = __builtin_amdgcn_wmma_f32_16x16x4_f32(
                      false, a, false, b, (short)0, acc, false, false);
        }
        // D layout: VGPR r -> M = mbase + r + (lane<16 ? 0 : 8), N = n.
        #pragma unroll
        for (int r = 0; r < 8; ++r) {
            int m = row0 + mbase + r + ((lane < 16) ? 0 : 8);
            if (m < N_NODES)
                out[(long)m * D + nbase + n] = acc[r] + bval;
        }
    }
}

// ---------------------------------------------------------------------------
extern "C" void kernel_launch(void* const* d_in, const int* in_sizes, int n_in,
                              void* d_out, int out_size, void* d_ws, size_t ws_size,
                              hipStream_t stream) {
    const float* feat = (const float*)d_in[0];  // [100000,128] f32
    const int*   src  = (const int*)  d_in[1];  // [1600000] i32
    const int*   dst  = (const int*)  d_in[2];  // [1600000] i32
    const float* W    = (const float*)d_in[3];  // [128,128] f32
    const float* bias = (const float*)d_in[4];  // [128] f32
    float* out = (float*)d_out;                 // [100000,128] f32
    float* h   = (float*)d_ws;                  // accumulator: 51.2 MB scratch

    // 1) zero h accumulator
    int n4 = (N_NODES * D) / 4;
    gc_zero_kernel<<<1024, 256, 0, stream>>>((float4*)d_ws, n4);

    // 2) edge scatter-add (1 wave per edge)
    long total_threads = (long)N_EDGES * 32;
    int  sblocks = (int)((total_threads + 255) / 256);
    gc_scatter_kernel<<<sblocks, 256, 0, stream>>>(feat, src, dst, h);

    // 3) WMMA f32 GEMM + bias
    int gblocks = (N_NODES + 63) / 64;
    gc_gemm_kernel<<<gblocks, 256, 0, stream>>>(h, W, bias, out);
}